// RobustCombiner_81982335746346
// MI455X (gfx1250) — compile-verified
//
#include <hip/hip_runtime.h>

// Problem constants (match reference)
#define BB 4
#define SS 1024
#define KK 32
#define VV 32000
#define MID 32
#define ROWS_PER_BLK 16   // one WMMA M-tile per block
#define THREADS 512       // 16 waves (wave32) -> one wave per row

typedef __attribute__((ext_vector_type(2))) float v2f;
typedef __attribute__((ext_vector_type(4))) float f32x4;
typedef __attribute__((ext_vector_type(8))) float v8f;

__device__ __forceinline__ float wave_sum32(float v) {
#pragma unroll
    for (int m = 16; m >= 1; m >>= 1) v += __shfl_xor(v, m, 32);
    return v;
}
__device__ __forceinline__ float wave_max32(float v) {
#pragma unroll
    for (int m = 16; m >= 1; m >>= 1) v = fmaxf(v, __shfl_xor(v, m, 32));
    return v;
}

__global__ __launch_bounds__(THREADS)
void robust_combiner_kernel(const int*   __restrict__ tgt_index,           // [B,S,K]
                            const float* __restrict__ knn_dists,           // [B,S,K] sorted
                            const float* __restrict__ knn_key_feature,     // [B,S,K]
                            const float* __restrict__ network_select_probs,// [B,S,K]
                            const float* __restrict__ fc1_w1,              // [2,4]
                            const float* __restrict__ fc1_b1,              // [4]
                            const float* __restrict__ fc1_w2,              // [4,1]
                            const float* __restrict__ fc1_b2,              // [1]
                            const float* __restrict__ fc2_w1,              // [64,32]
                            const float* __restrict__ fc2_b1,              // [32]
                            const float* __restrict__ fc2_w2,              // [32,2]
                            const float* __restrict__ fc2_b2,              // [2]
                            float*       __restrict__ out)                 // [B,S,V]
{
    __shared__ float s_feat[ROWS_PER_BLK][2 * KK]; // knn_feat staged for WMMA A
    __shared__ float s_h[ROWS_PER_BLK][MID];       // hidden = feat @ fc2_w1
    __shared__ int   s_tgt[ROWS_PER_BLK][KK];
    __shared__ int   s_if[ROWS_PER_BLK][KK];
    __shared__ float s_prob[ROWS_PER_BLK][KK];

    const int tid  = threadIdx.x;
    const int wave = tid >> 5;     // 0..15, one row per wave
    const int lane = tid & 31;     // 0..31, one neighbor per lane
    const int row  = blockIdx.x * ROWS_PER_BLK + wave;   // < B*S

    // ---------------------------------------------------------------
    // Phase 0: stream-zero this block's 16 output rows (NT 128b stores)
    // 16 * 32000 floats = 2 MB per block; no reuse -> non-temporal.
    // ---------------------------------------------------------------
    {
        f32x4* blk_out = (f32x4*)(out + (size_t)blockIdx.x * ROWS_PER_BLK * VV);
        const int nvec4 = ROWS_PER_BLK * VV / 4;  // 128000
        const f32x4 z = {0.f, 0.f, 0.f, 0.f};
        for (int i = tid; i < nvec4; i += THREADS)
            __builtin_nontemporal_store(z, blk_out + i);
    }

    // ---------------------------------------------------------------
    // Phase 1: per-row (per-wave) scalar path. Dead code (top-k over
    // network_probs, dfc_*, lambda_logit[...,0]) is eliminated: the
    // output depends only on tempe + noise_logit + knn_dists + tgt.
    // ---------------------------------------------------------------
    const size_t kbase = (size_t)row * KK + lane;
    const float d   = knn_dists[kbase];
    const float kf  = knn_key_feature[kbase];
    const float sp  = network_select_probs[kbase];
    const int   tg  = tgt_index[kbase];

    s_tgt[wave][lane] = tg;
    // first-occurrence flag (any value, for scatter dedup)
    bool dupBefore = false;
    for (int m = 0; m < lane; ++m) dupBefore |= (s_tgt[wave][m] == tg);
    // distinct-nonzero prefix counts via prefix sum of first-occurrence-of-nonzero
    s_if[wave][lane] = (tg != 0 && !dupBefore) ? 1 : 0;
    int cnt = 0;
    for (int j = 0; j <= lane; ++j) cnt += s_if[wave][j];

    // noise_logit: tanh([log kf, log sp] @ fc1_w1 + b1) @ fc1_w2 + b2
    const float lkf = logf(kf);
    const float lsp = logf(sp);
    float nl = fc1_b2[0];
#pragma unroll
    for (int j = 0; j < 4; ++j) {
        const float hj = tanhf(lkf * fc1_w1[j] + lsp * fc1_w1[4 + j] + fc1_b1[j]);
        nl += hj * fc1_w2[j];
    }

    // stage knn_feat = [dists | label_counts] for the WMMA tile
    s_feat[wave][lane]      = d;
    s_feat[wave][KK + lane] = (float)cnt;
    __syncthreads();

    // ---------------------------------------------------------------
    // Phase 2: fc2 first layer as f32 WMMA on wave 0:
    //   H[16x32] = feat[16x64] x fc2_w1[64x32]
    // 16 K-steps of V_WMMA_F32_16X16X4_F32 per N-tile, 2 N-tiles.
    // Layouts per ISA 7.12.2 (16x4 A, 4x16 B, 16x16 C/D).
    // ---------------------------------------------------------------
    if (wave == 0) {
        const int m  = lane & 15;
        const int hi = lane >> 4;       // 0: K pair {0,1}, 1: K pair {2,3}
        v8f acc0 = {0.f,0.f,0.f,0.f,0.f,0.f,0.f,0.f};
        v8f acc1 = {0.f,0.f,0.f,0.f,0.f,0.f,0.f,0.f};
#pragma unroll
        for (int kk = 0; kk < 16; ++kk) {
            const int k0 = kk * 4 + hi * 2;
            v2f a;  a.x  = s_feat[m][k0];
                    a.y  = s_feat[m][k0 + 1];
            v2f b0; b0.x = fc2_w1[(size_t)k0 * MID + m];
                    b0.y = fc2_w1[(size_t)(k0 + 1) * MID + m];
            v2f b1; b1.x = fc2_w1[(size_t)k0 * MID + 16 + m];
                    b1.y = fc2_w1[(size_t)(k0 + 1) * MID + 16 + m];
            acc0 = __builtin_amdgcn_wmma_f32_16x16x4_f32(
                false, a, false, b0, (short)0, acc0, false, false);
            acc1 = __builtin_amdgcn_wmma_f32_16x16x4_f32(
                false, a, false, b1, (short)0, acc1, false, false);
        }
#pragma unroll
        for (int r = 0; r < 8; ++r) {
            const int mr = r + hi * 8;       // D: VGPR r holds M=r (lanes 0-15) / M=r+8
            s_h[mr][m]      = acc0[r];
            s_h[mr][16 + m] = acc1[r];
        }
    }
    __syncthreads();

    // ---------------------------------------------------------------
    // Phase 3: tempe = sigmoid(channel 1 of fc2), softmax over K=32
    // ---------------------------------------------------------------
    const float t = tanhf(s_h[wave][lane] + fc2_b1[lane]);
    const float lambda1 = wave_sum32(t * fc2_w2[lane * 2 + 1]) + fc2_b2[1];
    const float tempe = 1.0f / (1.0f + expf(-lambda1));

    const float logit = -d * tempe + nl;
    const float mx = wave_max32(logit);
    const float e  = expf(logit - mx);
    const float p  = e / wave_sum32(e);
    s_prob[wave][lane] = p;

    // in-wave dedup of duplicate vocab indices (so scatter needs no atomics)
    float total = 0.f;
    for (int j = 0; j < KK; ++j)
        total += (s_tgt[wave][j] == tg) ? s_prob[wave][j] : 0.f;

    // ---------------------------------------------------------------
    // Phase 4: order zero-fill before scatter, then overwrite <=32 slots
    // (block exclusively owns its 16 rows -> plain stores are safe).
    // ---------------------------------------------------------------
    __threadfence();
    __syncthreads();

    if (!dupBefore)
        out[(size_t)row * VV + tg] = total;
}

extern "C" void kernel_launch(void* const* d_in, const int* in_sizes, int n_in,
                              void* d_out, int out_size, void* d_ws, size_t ws_size,
                              hipStream_t stream) {
    (void)in_sizes; (void)n_in; (void)d_ws; (void)ws_size; (void)out_size;
    // setup_inputs() order:
    // 0 tgt_index(int) 1 knn_dists 2 knn_key_feature 3 network_probs(DEAD)
    // 4 network_select_probs 5 dfc_w(DEAD) 6 dfc_b(DEAD)
    // 7 fc1_w1 8 fc1_b1 9 fc1_w2 10 fc1_b2 11 fc2_w1 12 fc2_b1 13 fc2_w2 14 fc2_b2
    const int*   tgt   = (const int*)  d_in[0];
    const float* dists = (const float*)d_in[1];
    const float* kfeat = (const float*)d_in[2];
    const float* nsp   = (const float*)d_in[4];
    const float* f1w1  = (const float*)d_in[7];
    const float* f1b1  = (const float*)d_in[8];
    const float* f1w2  = (const float*)d_in[9];
    const float* f1b2  = (const float*)d_in[10];
    const float* f2w1  = (const float*)d_in[11];
    const float* f2b1  = (const float*)d_in[12];
    const float* f2w2  = (const float*)d_in[13];
    const float* f2b2  = (const float*)d_in[14];
    float* outp = (float*)d_out;

    const int nBlocks = (BB * SS) / ROWS_PER_BLK;   // 256
    robust_combiner_kernel<<<nBlocks, THREADS, 0, stream>>>(
        tgt, dists, kfeat, nsp,
        f1w1, f1b1, f1w2, f1b2,
        f2w1, f2b1, f2w2, f2b2, outp);
}